// TaskAlignedAssigner_30562987278848
// MI455X (gfx1250) — compile-verified
//
#include <hip/hip_runtime.h>
#include <math.h>

#define NA    8400
#define NC    80
#define NMAX  64
#define BS    16
#define TOPK_ 13
#define EPSF  1e-9f
#define PI_F  3.14159265358979323846f

typedef float v2f __attribute__((ext_vector_type(2)));
typedef float v8f __attribute__((ext_vector_type(8)));

// ---------------------------------------------------------------------------
// K1: per (b, gt, anchor) alignment metric + circle IoU.
// WMMA f32 16x16x4 computes the rank-2 cross terms g.a and g.p for a
// 16(gt) x 16(anchor) tile per wave; 8 waves per block -> 16 gt x 128 anchors.
// d^2 = |g|^2 - 2*cross + |p|^2 (exact f32, same as scalar).
// ---------------------------------------------------------------------------
__global__ __launch_bounds__(256) void k1_metrics(
    const float* __restrict__ pd_scores,   // (BS, NA, NC)
    const float* __restrict__ pd_circles,  // (BS, NA, 3)
    const float* __restrict__ anc_points,  // (NA, 2)
    const int*   __restrict__ gt_labels,   // (BS, NMAX)
    const float* __restrict__ gt_bboxes,   // (BS, NMAX, 3)
    const float* __restrict__ mask_gt,     // (BS, NMAX)
    float* __restrict__ Aout,              // (BS, NMAX, NA) align metric (masked)
    float* __restrict__ Oout)              // (BS, NMAX, NA) overlaps (masked)
{
    const int b    = blockIdx.z;
    const int g0   = blockIdx.y * 16;
    const int a0   = blockIdx.x * 128;
    const int tid  = threadIdx.x;
    const int lane = tid & 31;
    const int wv   = tid >> 5;          // wave id (wave32)

    __shared__ float sgx[16], sgy[16], sgr[16], sgn[16], sgv[16];
    __shared__ int   sgl[16];
    if (tid < 16) {
        int g = g0 + tid;
        float gx = gt_bboxes[(b * NMAX + g) * 3 + 0];
        float gy = gt_bboxes[(b * NMAX + g) * 3 + 1];
        float gr = gt_bboxes[(b * NMAX + g) * 3 + 2];
        sgx[tid] = gx; sgy[tid] = gy; sgr[tid] = gr;
        sgn[tid] = gx * gx + gy * gy;
        sgv[tid] = mask_gt[b * NMAX + g];
        int lab  = gt_labels[b * NMAX + g];
        sgl[tid] = min(max(lab, 0), NC - 1);
    }
    __syncthreads();

    const int col = lane & 15;          // N (anchor column) / M row for lanes<16
    const int hi  = lane >> 4;          // lane half: supplies K=0,1 (lo) or K=2,3 (hi, zero-padded)
    const int n   = a0 + wv * 16 + col; // this lane's anchor column
    const int ncl = n < NA ? n : NA - 1;

    const float ax = anc_points[ncl * 2 + 0];
    const float ay = anc_points[ncl * 2 + 1];
    const float px = pd_circles[(b * NA + ncl) * 3 + 0];
    const float py = pd_circles[(b * NA + ncl) * 3 + 1];
    const float pr = pd_circles[(b * NA + ncl) * 3 + 2];

    // A: 16x4 gt centers (K=2,3 zero).  B: 4x16 anchor / predicted centers.
    v2f amat, b1, b2;
    if (hi == 0) {
        amat[0] = sgx[col]; amat[1] = sgy[col];
        b1[0] = ax; b1[1] = ay;
        b2[0] = px; b2[1] = py;
    } else {
        amat[0] = 0.f; amat[1] = 0.f;
        b1[0] = 0.f; b1[1] = 0.f;
        b2[0] = 0.f; b2[1] = 0.f;
    }
    v8f cz = {};
    v8f cross_a = __builtin_amdgcn_wmma_f32_16x16x4_f32(false, amat, false, b1,
                                                        (short)0, cz, false, false);
    v8f cross_p = __builtin_amdgcn_wmma_f32_16x16x4_f32(false, amat, false, b2,
                                                        (short)0, cz, false, false);

    const float anorm = ax * ax + ay * ay;
    const float pnorm = px * px + py * py;

#pragma unroll
    for (int r = 0; r < 8; ++r) {
        const int   m     = r + hi * 8;       // gt row within tile
        const float gn    = sgn[m];
        const float r1    = sgr[m];
        const float valid = sgv[m];
        const int   lab   = sgl[m];

        // anchor-in-gt mask: |g-a|^2 <= r1^2
        float dA2    = fmaxf(gn - 2.0f * cross_a[r] + anorm, 0.0f);
        float in_gts = (dA2 <= r1 * r1) ? 1.0f : 0.0f;
        float mv     = in_gts * valid;

        // circle IoU(gt, pred)
        float dd   = fmaxf(gn - 2.0f * cross_p[r] + pnorm, 0.0f);
        float d    = sqrtf(dd + EPSF);
        float d2   = d * d;
        float r2   = pr;
        float a1   = PI_F * r1 * r1;
        float a2   = PI_F * r2 * r2;
        float cos1 = fminf(fmaxf((d2 + r1 * r1 - r2 * r2) / (2.0f * d * r1 + EPSF), -1.0f), 1.0f);
        float cos2 = fminf(fmaxf((d2 + r2 * r2 - r1 * r1) / (2.0f * d * r2 + EPSF), -1.0f), 1.0f);
        float tri  = fmaxf((r1 + r2 - d) * (d + r1 - r2) * (d - r1 + r2) * (d + r1 + r2), 0.0f);
        float lens = r1 * r1 * acosf(cos1) + r2 * r2 * acosf(cos2) - 0.5f * sqrtf(tri);
        float rmin = fminf(r1, r2);
        float inter = (d >= r1 + r2) ? 0.0f
                      : ((d <= fabsf(r1 - r2)) ? PI_F * rmin * rmin : lens);
        float iou = inter / (a1 + a2 - inter + EPSF);

        float ov  = iou * mv;
        float sc  = pd_scores[((size_t)(b * NA + ncl)) * NC + lab] * mv;
        float o2  = ov * ov;
        float align = sc * o2 * o2 * o2;     // score^1 * iou^6 (masked)

        if (n < NA) {
            size_t off = ((size_t)(b * NMAX + g0 + m)) * NA + n;
            Aout[off] = align;
            Oout[off] = ov;
        }
    }
}

// ---------------------------------------------------------------------------
// K2: stable top-13 per (b, gt) row with in-circle re-check -> mask_pos.
// One block per row; align row cached in LDS (33.6 KB of the 320 KB WGP LDS).
// Tie-break: larger value, then smaller index (== jax.lax.top_k).
// ---------------------------------------------------------------------------
__global__ __launch_bounds__(256) void k2_topk(
    const float* __restrict__ anc_points,
    const float* __restrict__ gt_bboxes,   // (BS, NMAX, 3)
    const float* __restrict__ mask_gt,     // (BS, NMAX)
    const float* __restrict__ Ain,         // (BS, NMAX, NA)
    float* __restrict__ Mpos)              // (BS, NMAX, NA)
{
    const int bg  = blockIdx.x;            // b*NMAX + g
    const int tid = threadIdx.x;
    __shared__ float sRow[NA];
    __shared__ float sV[256];
    __shared__ int   sI[256];

    const size_t base = (size_t)bg * NA;
    for (int i = tid; i < NA; i += 256) {
        Mpos[base + i] = 0.0f;
        sRow[i] = Ain[base + i];
    }
    __syncthreads();

    const float gvalid = mask_gt[bg];
    if (gvalid <= 0.0f) return;            // uniform per block: invalid gt -> empty row

    const float gx = gt_bboxes[bg * 3 + 0];
    const float gy = gt_bboxes[bg * 3 + 1];
    const float gr = gt_bboxes[bg * 3 + 2];

    for (int k = 0; k < TOPK_; ++k) {
        float bv = -2.0f; int bi = 0x7fffffff;
        for (int i = tid; i < NA; i += 256) {
            float v = sRow[i];
            if (v > bv || (v == bv && i < bi)) { bv = v; bi = i; }
        }
        sV[tid] = bv; sI[tid] = bi;
        __syncthreads();
        for (int s = 128; s > 0; s >>= 1) {
            if (tid < s) {
                float v = sV[tid + s]; int i2 = sI[tid + s];
                if (v > sV[tid] || (v == sV[tid] && i2 < sI[tid])) { sV[tid] = v; sI[tid] = i2; }
            }
            __syncthreads();
        }
        if (tid == 0) {
            int w = sI[0];
            sRow[w] = -1.0f;               // exclude from next pass (align >= 0)
            float dx = gx - anc_points[w * 2 + 0];
            float dy = gy - anc_points[w * 2 + 1];
            if (dx * dx + dy * dy <= gr * gr) Mpos[base + w] = 1.0f;  // topk & in_gts & valid
        }
        __syncthreads();
    }
}

// ---------------------------------------------------------------------------
// K3: per (b, anchor) resolve multi-GT anchors by max overlap (first-max),
// finalize mask_pos, emit labels / circles / fg_mask / target_gt_idx.
// ---------------------------------------------------------------------------
__global__ __launch_bounds__(256) void k3_resolve(
    const int*   __restrict__ gt_labels,   // (BS, NMAX)
    const float* __restrict__ gt_bboxes,   // (BS, NMAX, 3)
    const float* __restrict__ Oin,         // (BS, NMAX, NA)
    float* __restrict__ Mpos,              // (BS, NMAX, NA) in/out
    float* __restrict__ out_labels,        // (BS, NA)
    float* __restrict__ out_circles,       // (BS, NA, 3)
    float* __restrict__ out_fg,            // (BS, NA)
    float* __restrict__ out_idx)           // (BS, NA)
{
    const int t = blockIdx.x * 256 + threadIdx.x;
    if (t >= BS * NA) return;
    const int b = t / NA;
    const int a = t - b * NA;
    const size_t base = (size_t)b * NMAX * NA + a;

    float cnt = 0.0f;
    for (int g = 0; g < NMAX; ++g) cnt += Mpos[base + (size_t)g * NA];

    float fg; int gi;
    if (cnt > 1.0f) {
        float bo = -1.0f; int bgm = 0;
        for (int g = 0; g < NMAX; ++g) {
            float o = Oin[base + (size_t)g * NA];
            if (o > bo) { bo = o; bgm = g; }   // strict > keeps first max (argmax)
        }
        for (int g = 0; g < NMAX; ++g)
            Mpos[base + (size_t)g * NA] = (g == bgm) ? 1.0f : 0.0f;
        fg = 1.0f; gi = bgm;
    } else {
        fg = cnt;
        gi = 0;
        for (int g = 0; g < NMAX; ++g) {
            if (Mpos[base + (size_t)g * NA] > 0.0f) { gi = g; break; }
        }
    }

    int lab = gt_labels[b * NMAX + gi];
    if (lab < 0) lab = 0;                       // clip(.., 0, None)
    out_labels[t] = (float)lab;
    out_circles[(size_t)t * 3 + 0] = gt_bboxes[(b * NMAX + gi) * 3 + 0];
    out_circles[(size_t)t * 3 + 1] = gt_bboxes[(b * NMAX + gi) * 3 + 1];
    out_circles[(size_t)t * 3 + 2] = gt_bboxes[(b * NMAX + gi) * 3 + 2];
    out_fg[t]  = fg;
    out_idx[t] = (float)gi;
}

// ---------------------------------------------------------------------------
// K4: per (b, gt) max over anchors of align*mask and overlaps*mask.
// ---------------------------------------------------------------------------
__global__ __launch_bounds__(256) void k4_pos(
    const float* __restrict__ Ain,
    const float* __restrict__ Oin,
    const float* __restrict__ Mpos,
    float* __restrict__ posA,              // (BS*NMAX)
    float* __restrict__ posO)              // (BS*NMAX)
{
    const int bg  = blockIdx.x;
    const int tid = threadIdx.x;
    const size_t base = (size_t)bg * NA;
    float ma = 0.0f, mo = 0.0f;
    for (int i = tid; i < NA; i += 256) {
        float m = Mpos[base + i];
        ma = fmaxf(ma, Ain[base + i] * m);
        mo = fmaxf(mo, Oin[base + i] * m);
    }
    __shared__ float sA[256], sO[256];
    sA[tid] = ma; sO[tid] = mo;
    __syncthreads();
    for (int s = 128; s > 0; s >>= 1) {
        if (tid < s) {
            sA[tid] = fmaxf(sA[tid], sA[tid + s]);
            sO[tid] = fmaxf(sO[tid], sO[tid + s]);
        }
        __syncthreads();
    }
    if (tid == 0) { posA[bg] = sA[0]; posO[bg] = sO[0]; }
}

// ---------------------------------------------------------------------------
// K5: normalized one-hot target scores. After resolution mask_pos has at most
// one active gt per anchor, so norm reduces to a single gather.
// ---------------------------------------------------------------------------
__global__ __launch_bounds__(256) void k5_scores(
    const float* __restrict__ Ain,
    const float* __restrict__ posA,
    const float* __restrict__ posO,
    const float* __restrict__ out_labels,
    const float* __restrict__ out_fg,
    const float* __restrict__ out_idx,
    float* __restrict__ out_scores)        // (BS, NA, NC)
{
    const int t = blockIdx.x * 256 + threadIdx.x;
    if (t >= BS * NA) return;
    const int b = t / NA;
    const int a = t - b * NA;

    const float fg = out_fg[t];
    float norm = 0.0f;
    int lab = 0;
    if (fg > 0.0f) {
        const int bg = b * NMAX + (int)out_idx[t];
        const float al = Ain[(size_t)bg * NA + a];   // == align*mask_pos at the active gt
        norm = al * posO[bg] / (posA[bg] + EPSF);
        lab  = (int)out_labels[t];
    }
    float* dst = out_scores + (size_t)t * NC;
    for (int c = 0; c < NC; ++c)
        dst[c] = (fg > 0.0f && c == lab) ? norm : 0.0f;
}

// ---------------------------------------------------------------------------
extern "C" void kernel_launch(void* const* d_in, const int* in_sizes, int n_in,
                              void* d_out, int out_size, void* d_ws, size_t ws_size,
                              hipStream_t stream) {
    const float* pd_scores  = (const float*)d_in[0];   // (16, 8400, 80)
    const float* pd_circles = (const float*)d_in[1];   // (16, 8400, 3)
    const float* anc        = (const float*)d_in[2];   // (8400, 2)
    const int*   gt_labels  = (const int*)  d_in[3];   // (16, 64, 1)
    const float* gt_bboxes  = (const float*)d_in[4];   // (16, 64, 3)
    const float* mask_gt    = (const float*)d_in[5];   // (16, 64, 1)

    float* out = (float*)d_out;   // outputs concatenated flat, return order
    float* out_labels  = out;                                   // (16, 8400)
    float* out_circles = out_labels + (size_t)BS * NA;          // (16, 8400, 3)
    float* out_scores  = out_circles + (size_t)BS * NA * 3;     // (16, 8400, 80)
    float* out_fg      = out_scores + (size_t)BS * NA * NC;     // (16, 8400)
    float* out_idx     = out_fg + (size_t)BS * NA;              // (16, 8400)

    const size_t CUBE = (size_t)BS * NMAX * NA;
    float* A    = (float*)d_ws;          // align metric
    float* O    = A + CUBE;              // overlaps
    float* M    = O + CUBE;              // mask_pos
    float* posA = M + CUBE;              // (BS*NMAX)
    float* posO = posA + BS * NMAX;

    dim3 g1((NA + 127) / 128, NMAX / 16, BS);
    k1_metrics<<<g1, 256, 0, stream>>>(pd_scores, pd_circles, anc, gt_labels,
                                       gt_bboxes, mask_gt, A, O);
    k2_topk<<<BS * NMAX, 256, 0, stream>>>(anc, gt_bboxes, mask_gt, A, M);
    k3_resolve<<<(BS * NA + 255) / 256, 256, 0, stream>>>(gt_labels, gt_bboxes, O, M,
                                                          out_labels, out_circles,
                                                          out_fg, out_idx);
    k4_pos<<<BS * NMAX, 256, 0, stream>>>(A, O, M, posA, posO);
    k5_scores<<<(BS * NA + 255) / 256, 256, 0, stream>>>(A, posA, posO, out_labels,
                                                         out_fg, out_idx, out_scores);
}